// GegEmbedding_40080634806647
// MI455X (gfx1250) — compile-verified
//
#include <hip/hip_runtime.h>

#define DIM_OUT 64
#define ALPHA   0.5f

#define TPB     128              // 4 waves of 32 (wave32 on gfx1250)
#define WAVES   (TPB / 32)
#define RSTRIDE 68               // 64 + 4 pad floats: 272 B row stride, 16B-aligned,
                                 // breaks the stride-64 all-lanes-same-bank pathology

#if defined(__has_builtin)
#if __has_builtin(__builtin_amdgcn_global_store_async_from_lds_b128)
#define HAVE_ASYNC_STORE 1
#endif
#endif

// The async builtin takes typed 16-byte-vector pointers (per round-1 diagnostic).
typedef int v4i __attribute__((ext_vector_type(4)));
typedef __attribute__((address_space(1))) v4i gas1_v4i;   // global
typedef __attribute__((address_space(3))) v4i las3_v4i;   // LDS

__global__ __launch_bounds__(TPB) void GegEmbedding_kernel(
    const float* __restrict__ xin, float* __restrict__ out, int n) {
  __shared__ float lds[WAVES * 32 * RSTRIDE];

  const int tid  = threadIdx.x;
  const int wave = tid >> 5;
  const int lane = tid & 31;
  const int row  = blockIdx.x * TPB + tid;

  float* wlds = &lds[wave * 32 * RSTRIDE];   // this wave's 32-row tile
  float* my   = wlds + lane * RSTRIDE;       // this lane's row (272 B stride)

  float x = 0.0f;
  if (row < n) x = xin[row];                 // coalesced b32 loads

  // C_1 = 2*alpha*x ; C_0 = 1
  float prev  = 2.0f * ALPHA * x;
  float prev2 = 1.0f;
  my[0] = prev;

  // Fully unrolled: A,B constant-fold (exact f32 divides at compile time),
  // so the loop body is mul + mul + fma — no runtime division.
#pragma unroll
  for (int i = 2; i <= DIM_OUT; ++i) {
    const float A = (2.0f * (float)i - 2.0f + 2.0f * ALPHA) / (float)i;
    const float B = (-(float)i + 2.0f - 2.0f * ALPHA) / (float)i;
    float ci = __builtin_fmaf(A * x, prev, B * prev2);
    my[i - 1] = ci;
    prev2 = prev;
    prev  = ci;
  }

  const long long waveBase = (long long)blockIdx.x * TPB + (long long)(wave * 32);
  const bool fullTile = (waveBase + 32) <= (long long)n;

#ifdef HAVE_ASYNC_STORE
  if (fullTile) {
    // Drain this wave's LDS writes before the async DMA reads them
    // (DS ops tracked by DScnt; async LDS->global tracked by ASYNCcnt).
    asm volatile("s_wait_dscnt 0x0" ::: "memory");

    float* gbase = out + (size_t)waveBase * DIM_OUT;   // 8 KB contiguous tile
    // 16 iterations x 32 lanes x 16 B = fully coalesced 512 B per instruction,
    // LDS -> global with no VGPR round-trip. S_ENDPGM's implicit wait-idle
    // covers ASYNCcnt completion.
#pragma unroll
    for (int j = 0; j < 16; ++j) {
      const int f = j * 32 + lane;        // float4 index within tile, 0..511
      const int r = f >> 4;               // tile row 0..31
      const int c = f & 15;               // float4 column 0..15
      __builtin_amdgcn_global_store_async_from_lds_b128(
          (gas1_v4i*)(gbase + (size_t)f * 4),
          (las3_v4i*)(wlds + r * RSTRIDE + c * 4),
          /*offset=*/0, /*cpol=*/0);
    }
    return;
  }
#endif

  // Fallback / partial-tile path: per-row vectorized b128 stores from LDS.
  if (row < n) {
    float4*       dst = (float4*)(out + (size_t)row * DIM_OUT);
    const float4* src = (const float4*)my;   // 272 B stride keeps 16B alignment
#pragma unroll
    for (int k = 0; k < DIM_OUT / 4; ++k) dst[k] = src[k];
  }
}

extern "C" void kernel_launch(void* const* d_in, const int* in_sizes, int n_in,
                              void* d_out, int out_size, void* d_ws, size_t ws_size,
                              hipStream_t stream) {
  const float* x   = (const float*)d_in[0];
  float*       out = (float*)d_out;
  const int    n   = in_sizes[0];          // 2,000,000 rows
  const int    blocks = (n + TPB - 1) / TPB;
  hipLaunchKernelGGL(GegEmbedding_kernel, dim3(blocks), dim3(TPB), 0, stream,
                     x, out, n);
}